// TempGCN_37589553774842
// MI455X (gfx1250) — compile-verified
//
#include <hip/hip_runtime.h>
#include <math.h>

#define N 8192
#define C 256

typedef float v4f __attribute__((ext_vector_type(4)));
typedef float v2f __attribute__((ext_vector_type(2)));
typedef float v8f __attribute__((ext_vector_type(8)));

// ---------------------------------------------------------------------------
// Kernel 1: E_new = 0.5 * Ep, with row `head` fixup:
//   row != head : out = 0.5*E[row]
//   row == head : out = 0.5*E[head] + (head!=tail ? 0.5*E[tail] : 0) + 0.5*d(j==tail)
// Streaming: 256 MB in + 256 MB out, non-temporal B128 (E >> 192MB L2, touched once).
// ---------------------------------------------------------------------------
__global__ void k_scale_E(const float* __restrict__ E, float* __restrict__ Eout,
                          const int* __restrict__ headp, const int* __restrict__ tailp) {
  const int head = *headp;
  const int tail = *tailp;
  size_t g = (size_t)blockIdx.x * blockDim.x + threadIdx.x;  // float4 index
  size_t base = g * 4;                                       // element index
  int row = (int)(base >> 13);                               // / 8192

  const v4f* in  = (const v4f*)E + g;
  v4f*       out = (v4f*)Eout + g;
  v4f e = __builtin_nontemporal_load(in);

  if (row != head) {
    v4f r = e * 0.5f;
    __builtin_nontemporal_store(r, out);
  } else {
    int j0 = (int)(base & (N - 1));
    v4f t = {0.f, 0.f, 0.f, 0.f};
    if (head != tail) {
      const v4f* tin = (const v4f*)E + (size_t)tail * (N / 4) + (g & (N / 4 - 1));
      t = __builtin_nontemporal_load(tin);
    }
    v4f r;
#pragma unroll
    for (int i = 0; i < 4; ++i) {
      float add = ((j0 + i) == tail) ? 0.5f : 0.0f;
      r[i] = 0.5f * e[i] + 0.5f * t[i] + add;
    }
    __builtin_nontemporal_store(r, out);
  }
}

// ---------------------------------------------------------------------------
// Kernel 2: fused  (a) partial agg[c] = sum_i w[i]*H[i][c]  over this block's 32 rows
//                  (b) H_new = H  copy (H read exactly once)
// w = E_new[head] read from d_out (produced by kernel 1). 256 blocks x 256 threads.
// Deterministic fixed-order accumulation (no float atomics).
// ---------------------------------------------------------------------------
__global__ void k_aggH(const float* __restrict__ H, const float* __restrict__ Eout,
                       float* __restrict__ Hnew, float* __restrict__ partial,
                       const int* __restrict__ headp) {
  const int head = *headp;
  __shared__ float wsh[32];
  const int c  = threadIdx.x;
  const int r0 = blockIdx.x * 32;
  if (c < 32) wsh[c] = Eout[(size_t)head * N + r0 + c];
  __syncthreads();

  float acc = 0.f;
#pragma unroll 4
  for (int r = 0; r < 32; ++r) {
    size_t idx = (size_t)(r0 + r) * C + c;
    float hv = H[idx];
    acc = fmaf(wsh[r], hv, acc);
    Hnew[idx] = hv;
  }
  partial[(size_t)blockIdx.x * C + c] = acc;
}

// Kernel 3: agg[c] = sum_b partial[b][c], fixed ascending order (deterministic).
__global__ void k_reduce(const float* __restrict__ partial, float* __restrict__ agg) {
  const int c = threadIdx.x;
  float acc = 0.f;
  for (int b = 0; b < 256; ++b) acc += partial[(size_t)b * C + c];
  agg[c] = acc;
}

// ---------------------------------------------------------------------------
// WMMA helper: one K=4 step of an exact fp32 matvec via V_WMMA_F32_16X16X4_F32.
//   A (16x4): only row M=0 live -> lane 0 carries {v[kk],v[kk+1]} (K=0,1),
//             lane 16 carries K=2,3; all other lanes zero (single v_cndmask).
//   B (4x16): lane n / half h carries {W[kk][col+n], W[kk+1][col+n]}, kk=k+2h.
//   D row 0 lands in acc[0] of lanes 0..15.
// Segment/base-pointer selection is done with UNIFORM loop bounds so there is
// no divergent control flow anywhere in the WMMA loops (EXEC stays all-1s).
// ---------------------------------------------------------------------------
__device__ __forceinline__ v8f wmma_step(v8f acc, float a0, float a1,
                                         const float* __restrict__ W,
                                         int kk, int coln, bool row0) {
  a0 = row0 ? a0 : 0.f;
  a1 = row0 ? a1 : 0.f;
  float b0 = W[(size_t)kk * C + coln];
  float b1 = W[(size_t)(kk + 1) * C + coln];
  v2f a = {a0, a1};
  v2f b = {b0, b1};
  return __builtin_amdgcn_wmma_f32_16x16x4_f32(false, a, false, b, (short)0, acc,
                                               false, false);
}

// Z and R gates: pre_g = [X | h_head | agg] @ [W_x_g ; W_h_g ; conv_g] + b_g
__global__ void k_gates_zr(const float* __restrict__ X, const float* __restrict__ H,
                           const float* __restrict__ agg,
                           const float* __restrict__ Wxz, const float* __restrict__ Whz,
                           const float* __restrict__ Cz,  const float* __restrict__ bz,
                           const float* __restrict__ Wxr, const float* __restrict__ Whr,
                           const float* __restrict__ Cr,  const float* __restrict__ br,
                           float* __restrict__ Zout, float* __restrict__ Rout,
                           const int* __restrict__ headp) {
  const int head = *headp;
  const int gate = blockIdx.x >> 4;          // 0 = z, 1 = r
  const int col0 = (blockIdx.x & 15) * 16;
  const float* Wx = gate ? Wxr : Wxz;
  const float* Wh = gate ? Whr : Whz;
  const float* Cv = gate ? Cr  : Cz;
  const float* bb = gate ? br  : bz;
  float*       oo = gate ? Rout : Zout;
  const float* hrow = H + (size_t)head * C;

  const int  lane = threadIdx.x;
  const int  half = lane >> 4;
  const int  n    = lane & 15;
  const int  coln = col0 + n;
  const bool row0 = (n == 0);

  v8f acc = {};
  // Segment 0: X @ W_x_g   (uniform base pointers, no divergence)
  for (int k = 0; k < C; k += 4) {
    int kk = k + 2 * half;
    acc = wmma_step(acc, X[kk], X[kk + 1], Wx, kk, coln, row0);
  }
  // Segment 1: h_head @ W_h_g
  for (int k = 0; k < C; k += 4) {
    int kk = k + 2 * half;
    acc = wmma_step(acc, hrow[kk], hrow[kk + 1], Wh, kk, coln, row0);
  }
  // Segment 2: agg @ conv_g
  for (int k = 0; k < C; k += 4) {
    int kk = k + 2 * half;
    acc = wmma_step(acc, agg[kk], agg[kk + 1], Cv, kk, coln, row0);
  }

  if (lane < 16) {
    float pre = acc[0] + bb[coln];
    oo[coln] = 1.0f / (1.0f + expf(-pre));   // sigmoid
  }
}

// H_tilde = tanh([X | h_head*R] @ [W_x_h ; W_h_h] + b_h); GRU update of row head.
__global__ void k_gate_h(const float* __restrict__ X, const float* __restrict__ H,
                         const float* __restrict__ Wxh, const float* __restrict__ Whh,
                         const float* __restrict__ bh,
                         const float* __restrict__ Zv, const float* __restrict__ Rv,
                         float* __restrict__ Hnew, const int* __restrict__ headp) {
  const int head = *headp;
  const int col0 = blockIdx.x * 16;
  const float* hrow = H + (size_t)head * C;
  const int  lane = threadIdx.x;
  const int  half = lane >> 4;
  const int  n    = lane & 15;
  const int  coln = col0 + n;
  const bool row0 = (n == 0);

  v8f acc = {};
  // Segment 0: X @ W_x_h
  for (int k = 0; k < C; k += 4) {
    int kk = k + 2 * half;
    acc = wmma_step(acc, X[kk], X[kk + 1], Wxh, kk, coln, row0);
  }
  // Segment 1: (h_head * R) @ W_h_h
  for (int k = 0; k < C; k += 4) {
    int kk = k + 2 * half;
    float a0 = hrow[kk] * Rv[kk];
    float a1 = hrow[kk + 1] * Rv[kk + 1];
    acc = wmma_step(acc, a0, a1, Whh, kk, coln, row0);
  }

  if (lane < 16) {
    float ht = tanhf(acc[0] + bh[coln]);
    float z  = Zv[coln];
    float hh = hrow[coln];
    Hnew[(size_t)head * C + coln] = z * hh + (1.f - z) * ht;
  }
}

// ---------------------------------------------------------------------------
extern "C" void kernel_launch(void* const* d_in, const int* in_sizes, int n_in,
                              void* d_out, int out_size, void* d_ws, size_t ws_size,
                              hipStream_t stream) {
  const float* X     = (const float*)d_in[0];
  const int*   headp = (const int*)d_in[1];
  const int*   tailp = (const int*)d_in[2];
  const float* E     = (const float*)d_in[3];
  const float* H     = (const float*)d_in[4];
  const float* Wxz   = (const float*)d_in[5];
  const float* Whz   = (const float*)d_in[6];
  const float* Cz    = (const float*)d_in[7];
  const float* bz    = (const float*)d_in[8];
  const float* Wxr   = (const float*)d_in[9];
  const float* Whr   = (const float*)d_in[10];
  const float* Cr    = (const float*)d_in[11];
  const float* br    = (const float*)d_in[12];
  const float* Wxh   = (const float*)d_in[13];
  const float* Whh   = (const float*)d_in[14];
  const float* bh    = (const float*)d_in[15];

  float* Eout = (float*)d_out;                      // [N*N]
  float* Hnew = (float*)d_out + (size_t)N * N;      // [N*C]

  float* partial = (float*)d_ws;                    // 256*C floats
  float* agg     = partial + 256 * C;               // C floats
  float* Zv      = agg + C;                         // C floats
  float* Rv      = Zv + C;                          // C floats

  // 1) stream-scale E (head row fixed up inline): 16M float4 -> 65536 blocks
  k_scale_E<<<(N / 4) * (N / 256), 256, 0, stream>>>(E, Eout, headp, tailp);
  // 2) fused agg partials + H copy: 256 blocks x 256 threads, 32 rows each
  k_aggH<<<N / 32, C, 0, stream>>>(H, Eout, Hnew, partial, headp);
  // 3) deterministic reduce of partials
  k_reduce<<<1, C, 0, stream>>>(partial, agg);
  // 4) Z and R gates via fp32 WMMA (32 waves: 2 gates x 16 column tiles)
  k_gates_zr<<<32, 32, 0, stream>>>(X, H, agg, Wxz, Whz, Cz, bz,
                                    Wxr, Whr, Cr, br, Zv, Rv, headp);
  // 5) H_tilde + GRU update of row `head` via fp32 WMMA (16 waves)
  k_gate_h<<<16, 32, 0, stream>>>(X, H, Wxh, Whh, bh, Zv, Rv, Hnew, headp);
}